// Agent_61753039782225
// MI455X (gfx1250) — compile-verified
//
#include <hip/hip_runtime.h>
#include <math.h>

typedef __attribute__((ext_vector_type(2))) float v2f;
typedef __attribute__((ext_vector_type(8))) float v8f;

// ---------------------------------------------------------------------------
// 1) avg-pool x (16,3,512,512) -> 64x64 and build enriched (16,14,64,64)
// ---------------------------------------------------------------------------
__global__ void k_pool_enrich(const float* __restrict__ x,
                              const float* __restrict__ states,
                              float* __restrict__ enr) {
  int idx = blockIdx.x * 256 + threadIdx.x;
  const int total = 16 * 14 * 64 * 64;
  if (idx >= total) return;
  int px = idx & 63;
  int py = (idx >> 6) & 63;
  int c  = (idx >> 12) % 14;
  int b  = idx / (14 * 4096);
  float v;
  if (c < 3) {
    const float* src = x + (((size_t)(b * 3 + c) * 512 + py * 8) * 512 + px * 8);
    float s = 0.f;
#pragma unroll
    for (int r = 0; r < 8; ++r) {
      const float4* p = (const float4*)(src + r * 512);
      float4 a0 = p[0], a1 = p[1];
      s += a0.x + a0.y + a0.z + a0.w + a1.x + a1.y + a1.z + a1.w;
    }
    v = s * (1.0f / 64.0f);
  } else {
    v = states[b * 11 + (c - 3)];
  }
  enr[idx] = v;
}

// ---------------------------------------------------------------------------
// 2) fused implicit-GEMM conv (stride 2, pad 1, 4x4) via V_WMMA_F32_16X16X4_F32
//    One wave computes a 16x16 (Cout x pixel) tile; the im2col B-fragment is
//    gathered on the fly: lane's two K-values are adjacent columns of one
//    input row, so only 2 bounds-checked loads per wmma step.
//    out = leaky( gamma*(conv + bias)/sqrt(1+eps) + beta )
// ---------------------------------------------------------------------------
template <int Cin, int IH, int Cout, int OH>
__global__ void k_conv_wmma(const float* __restrict__ Wt, const float* __restrict__ bias,
                            const float* __restrict__ gamma, const float* __restrict__ beta,
                            const float* __restrict__ src, float* __restrict__ out) {
  constexpr int K = Cin * 16;
  constexpr int OHW = OH * OH;
  int lane = threadIdx.x;          // 0..31
  int tm = blockIdx.x;             // M tile (Cout/16)
  int tn = blockIdx.y;             // N tile (B*OH*OH/16)
  int half = lane >> 4;            // 0: K = k0,k0+1  1: K = k0+2,k0+3
  int l16 = lane & 15;

  // pixel this lane owns (B fragment column)
  int n = tn * 16 + l16;
  int bb = n / OHW;                // constexpr divisor -> shifts
  int p = n % OHW;
  int oy = p / OH, ox = p % OH;
  int ix = ox * 2 - 1 + 2 * half;  // kw = 2*half (even), elements at ix, ix+1

  const float* aP = Wt + (size_t)(tm * 16 + l16) * K + 2 * half;
  const float* sB = src + (size_t)bb * Cin * IH * IH;

  v8f acc = {};
  for (int ci = 0; ci < Cin; ++ci) {
    const float* plane = sB + ci * IH * IH;
    const float* aCi = aP + ci * 16;
#pragma unroll
    for (int kh = 0; kh < 4; ++kh) {
      int iy = oy * 2 - 1 + kh;
      float e0 = 0.f, e1 = 0.f;
      if ((unsigned)iy < (unsigned)IH) {
        const float* row = plane + iy * IH;
        if (ix >= 0) e0 = row[ix];
        if (ix + 1 < IH) e1 = row[ix + 1];
      }
      v2f bfrag = {e0, e1};
      v2f a = *(const v2f*)(aCi + kh * 4);
      acc = __builtin_amdgcn_wmma_f32_16x16x4_f32(false, a, false, bfrag,
                                                  (short)0, acc, false, false);
    }
  }

  const float bn = rsqrtf(1.0f + 1e-5f);
#pragma unroll
  for (int r2 = 0; r2 < 8; ++r2) {
    int m = tm * 16 + r2 + 8 * half;   // C/D layout: vgpr r -> M=r (+8 for hi lanes)
    float v = acc[r2] + bias[m];
    v = gamma[m] * v * bn + beta[m];
    v = v >= 0.f ? v : 0.2f * v;
    out[((size_t)(bb * Cout + m) * OH + oy) * OH + ox] = v;
  }
}

// ---------------------------------------------------------------------------
// 3) FC1: hidden(16,512) = leaky( feats(16,4096) @ fc1_W^T + fc1_b )  (WMMA)
// ---------------------------------------------------------------------------
__global__ void k_fc1_wmma(const float* __restrict__ feats, const float* __restrict__ W,
                           const float* __restrict__ bias, float* __restrict__ hidden) {
  int lane = threadIdx.x;
  int tn = blockIdx.x;             // 0..31
  int half = lane >> 4, l16 = lane & 15;
  const float* aP = feats + (size_t)l16 * 4096 + 2 * half;
  const float* bP = W + (size_t)(tn * 16 + l16) * 4096 + 2 * half;
  v8f acc = {};
  for (int k0 = 0; k0 < 4096; k0 += 4) {
    v2f a  = *(const v2f*)(aP + k0);
    v2f b2 = *(const v2f*)(bP + k0);
    acc = __builtin_amdgcn_wmma_f32_16x16x4_f32(false, a, false, b2,
                                                (short)0, acc, false, false);
  }
  int n = tn * 16 + l16;
#pragma unroll
  for (int r2 = 0; r2 < 8; ++r2) {
    int m = r2 + 8 * half;               // batch row
    float v = acc[r2] + bias[n];
    v = v >= 0.f ? v : 0.2f * v;
    hidden[m * 512 + n] = v;
  }
}

// ---------------------------------------------------------------------------
// 4) A = tanh(feats_fe @ filt_W^T + filt_b) : (16,8) block-reduced dots
// ---------------------------------------------------------------------------
__global__ void k_filt(const float* __restrict__ feats, const float* __restrict__ W,
                       const float* __restrict__ bias, float* __restrict__ A) {
  __shared__ float sh[256];
  int b = blockIdx.x, j = blockIdx.y, t = threadIdx.x;
  const float* f = feats + (size_t)b * 4096;
  const float* w = W + (size_t)j * 4096;
  float s = 0.f;
  for (int i = t; i < 4096; i += 256) s += f[i] * w[i];
  sh[t] = s;
  __syncthreads();
  for (int stride = 128; stride > 0; stride >>= 1) {
    if (t < stride) sh[t] += sh[t + stride];
    __syncthreads();
  }
  if (t == 0) A[b * 8 + j] = tanhf(sh[0] + bias[j]);
}

// ---------------------------------------------------------------------------
// 5) head: fc2 + softmax + pdf mix + entropy + argmax + new_states/surrogate,
//    stash per-batch scale=exp2(A[sel]) and scalar penalty terms; zero sumsq.
// ---------------------------------------------------------------------------
__global__ void k_head(const float* __restrict__ Ab, const float* __restrict__ hidden,
                       const float* __restrict__ W2, const float* __restrict__ b2,
                       const float* __restrict__ states, const float* __restrict__ progress,
                       const float* __restrict__ runtime,
                       float* __restrict__ scaleBuf, float* __restrict__ penBase,
                       float* __restrict__ sumsq, float* __restrict__ outNS,
                       float* __restrict__ outSur) {
  int b = threadIdx.x;
  if (b >= 16) return;
  float logit[8];
  for (int j = 0; j < 8; ++j) {
    float s = b2[j];
    const float* h = hidden + b * 512;
    const float* w = W2 + j * 512;
    for (int i = 0; i < 512; ++i) s += h[i] * w[i];
    logit[j] = s;
  }
  float mx = logit[0];
  for (int j = 1; j < 8; ++j) mx = fmaxf(mx, logit[j]);
  float e[8], se = 0.f;
  for (int j = 0; j < 8; ++j) { e[j] = expf(logit[j] - mx); se += e[j]; }
  float pdf[8], sp = 0.f;
  for (int j = 0; j < 8; ++j) {
    pdf[j] = e[j] / se + 1e-37f;
    pdf[j] = pdf[j] * (1.0f - 0.05f) + 0.05f / 8.0f;
    sp += pdf[j];
  }
  float inv = 1.f / (sp + 1e-30f);
  float ent = 0.f;
  int sel = 0;
  float best = -1.f;
  for (int j = 0; j < 8; ++j) {
    pdf[j] *= inv;
    ent += -pdf[j] * logf(pdf[j]);
    if (pdf[j] > best) { best = pdf[j]; sel = j; }  // first max, like argmax
  }
  float st2 = states[b * 11 + 2];
  float isLast = (fabsf(st2 + 1.f - 5.f) < 1e-4f) ? 1.f : 0.f;
  float submitted = isLast;
  float entPen = (1.f - progress[0]) * 0.05f * (-ent + logf(8.f));
  float early  = (1.f - isLast) * submitted;
  outNS[b * 11 + 0] = submitted;
  outNS[b * 11 + 1] = submitted;
  outNS[b * 11 + 2] = st2 + 1.f;
  for (int j = 0; j < 8; ++j)
    outNS[b * 11 + 3 + j] = fmaxf(states[b * 11 + 3 + j], (j == sel) ? 1.f : 0.f);
  outSur[b] = logf(pdf[sel] + 1e-10f);
  scaleBuf[b] = exp2f(Ab[b * 8 + sel]);
  penBase[b] = entPen + states[b * 11 + 3 + sel] * 1.0f + early + 0.01f * runtime[sel];
  sumsq[b] = 0.f;
}

// ---------------------------------------------------------------------------
// 6) bandwidth pass: x_out = clip(x*scale[b],0,5); accumulate sum(max(x-1,0)^2)
//    per batch. Each block owns 4096 contiguous elements (float4 IO).
// ---------------------------------------------------------------------------
__global__ void k_xout(const float* __restrict__ x, const float* __restrict__ scaleBuf,
                       float* __restrict__ out, float* __restrict__ sumsq) {
  __shared__ float sh[256];
  size_t base = (size_t)blockIdx.x * 4096;
  int b = (int)(base / 786432);  // 3*512*512 per batch element
  float sc = scaleBuf[b];
  const float4* xi = (const float4*)(x + base);
  float4* xo = (float4*)(out + base);
  int t = threadIdx.x;
  float s = 0.f;
#pragma unroll
  for (int i = 0; i < 4; ++i) {
    float4 v = xi[i * 256 + t];
    float4 r;
    r.x = fminf(fmaxf(v.x * sc, 0.f), 5.f);
    r.y = fminf(fmaxf(v.y * sc, 0.f), 5.f);
    r.z = fminf(fmaxf(v.z * sc, 0.f), 5.f);
    r.w = fminf(fmaxf(v.w * sc, 0.f), 5.f);
    xo[i * 256 + t] = r;
    float a0 = fmaxf(r.x - 1.f, 0.f), a1 = fmaxf(r.y - 1.f, 0.f);
    float a2 = fmaxf(r.z - 1.f, 0.f), a3 = fmaxf(r.w - 1.f, 0.f);
    s += a0 * a0 + a1 * a1 + a2 * a2 + a3 * a3;
  }
  sh[t] = s;
  __syncthreads();
  for (int stride = 128; stride > 0; stride >>= 1) {
    if (t < stride) sh[t] += sh[t + stride];
    __syncthreads();
  }
  if (t == 0) atomicAdd(&sumsq[b], sh[0]);
}

__global__ void k_final(const float* __restrict__ sumsq, const float* __restrict__ penBase,
                        float* __restrict__ outPen) {
  int b = threadIdx.x;
  if (b < 16) outPen[b] = sumsq[b] * (1.0f / 786432.0f) + penBase[b];
}

// ---------------------------------------------------------------------------
extern "C" void kernel_launch(void* const* d_in, const int* in_sizes, int n_in,
                              void* d_out, int out_size, void* d_ws, size_t ws_size,
                              hipStream_t stream) {
  (void)out_size; (void)ws_size;
  const float* in[64];
  for (int i = 0; i < n_in && i < 64; ++i) in[i] = (const float*)d_in[i];
  const float *x = in[0], *states = in[2], *progress = in[3], *runtime = in[4];
  const float *feW[4], *feB[4], *feG[4], *feBe[4];
  const float *selW[4], *selB[4], *selG[4], *selBe[4];
  const float *fc1W, *fc1b, *fc2W, *fc2b, *filtW, *filtb;

  if (n_in >= 43 && in_sizes[5] == 2097152) {
    // params flattened with sorted dict keys (jax tree_leaves):
    // fc1_W, fc1_b, fc2_W, fc2_b, fe[ W,b,beta,gamma ]x4, filt_W, filt_b, sel[...]x4
    fc1W = in[5]; fc1b = in[6]; fc2W = in[7]; fc2b = in[8];
    for (int l = 0; l < 4; ++l) {
      feW[l] = in[9 + 4 * l]; feB[l] = in[10 + 4 * l];
      feBe[l] = in[11 + 4 * l]; feG[l] = in[12 + 4 * l];
    }
    filtW = in[25]; filtb = in[26];
    for (int l = 0; l < 4; ++l) {
      selW[l] = in[27 + 4 * l]; selB[l] = in[28 + 4 * l];
      selBe[l] = in[29 + 4 * l]; selG[l] = in[30 + 4 * l];
    }
  } else {
    // params flattened in insertion order:
    // fe[ W,b,gamma,beta ]x4, sel[...]x4, fc1_W, fc1_b, fc2_W, fc2_b, filt_W, filt_b
    for (int l = 0; l < 4; ++l) {
      feW[l] = in[5 + 4 * l]; feB[l] = in[6 + 4 * l];
      feG[l] = in[7 + 4 * l]; feBe[l] = in[8 + 4 * l];
    }
    for (int l = 0; l < 4; ++l) {
      selW[l] = in[21 + 4 * l]; selB[l] = in[22 + 4 * l];
      selG[l] = in[23 + 4 * l]; selBe[l] = in[24 + 4 * l];
    }
    fc1W = in[37]; fc1b = in[38]; fc2W = in[39]; fc2b = in[40];
    filtW = in[41]; filtb = in[42];
  }

  // workspace layout (floats) -- no im2col scratch needed anymore
  float* ws = (float*)d_ws;
  size_t o = 0;
  float* enr      = ws + o; o += 917504;    // (16,14,64,64)
  float* actA     = ws + o; o += 524288;    // ping
  float* actB     = ws + o; o += 262144;    // pong
  float* featsFe  = ws + o; o += 65536;     // (16,4096)
  float* featsSel = ws + o; o += 65536;
  float* hidden   = ws + o; o += 8192;      // (16,512)
  float* Abuf     = ws + o; o += 128;       // (16,8)
  float* scaleBuf = ws + o; o += 16;
  float* penBase  = ws + o; o += 16;
  float* sumsq    = ws + o; o += 16;

  float* outX   = (float*)d_out;            // (16,3,512,512)
  float* outNS  = outX + 12582912;          // (16,11)
  float* outSur = outNS + 176;              // (16,1)
  float* outPen = outSur + 16;              // (16,1)

  // 1) pool + enrich
  k_pool_enrich<<<(917504 + 255) / 256, 256, 0, stream>>>(x, states, enr);

  // 2) two conv branches, 4 layers each, fused im2col + WMMA GEMM
  for (int br = 0; br < 2; ++br) {
    const float* const* Wp = br ? selW : feW;
    const float* const* Bp = br ? selB : feB;
    const float* const* Gp = br ? selG : feG;
    const float* const* Ep = br ? selBe : feBe;
    float* feats = br ? featsSel : featsFe;
    // L1: (14,64) -> (32,32); grid (Cout/16, B*OH*OH/16)
    k_conv_wmma<14, 64, 32, 32><<<dim3(2, 1024), 32, 0, stream>>>(
        Wp[0], Bp[0], Gp[0], Ep[0], enr, actA);
    // L2: (32,32) -> (64,16)
    k_conv_wmma<32, 32, 64, 16><<<dim3(4, 256), 32, 0, stream>>>(
        Wp[1], Bp[1], Gp[1], Ep[1], actA, actB);
    // L3: (64,16) -> (128,8)
    k_conv_wmma<64, 16, 128, 8><<<dim3(8, 64), 32, 0, stream>>>(
        Wp[2], Bp[2], Gp[2], Ep[2], actB, actA);
    // L4: (128,8) -> (256,4) == feats (16,4096)
    k_conv_wmma<128, 8, 256, 4><<<dim3(16, 16), 32, 0, stream>>>(
        Wp[3], Bp[3], Gp[3], Ep[3], actA, feats);
  }

  // 3) heads
  k_fc1_wmma<<<32, 32, 0, stream>>>(featsSel, fc1W, fc1b, hidden);
  k_filt<<<dim3(16, 8), 256, 0, stream>>>(featsFe, filtW, filtb, Abuf);
  k_head<<<1, 16, 0, stream>>>(Abuf, hidden, fc2W, fc2b, states, progress, runtime,
                               scaleBuf, penBase, sumsq, outNS, outSur);

  // 4) fused bandwidth pass + penalty reduction
  k_xout<<<3072, 256, 0, stream>>>(x, scaleBuf, outX, sumsq);
  k_final<<<1, 16, 0, stream>>>(sumsq, penBase, outPen);
}